// GCNLayer_17480516894879
// MI455X (gfx1250) — compile-verified
//
#include <hip/hip_runtime.h>

// ---------------------------------------------------------------------------
// GCN layer for MI455X (gfx1250, wave32).
//   gather -> atomic scatter-mean -> f32 WMMA GEMM (16x16x4) -> graphnorm
//   -> batchnorm (2-pass, column stats) -> relu + residual
// CDNA5 features: v_wmma_f32_16x16x4_f32, global_load_async_to_lds_b128.
// ---------------------------------------------------------------------------

typedef float v2f __attribute__((ext_vector_type(2)));
typedef float v8f __attribute__((ext_vector_type(8)));

#define DIM 128
#define HS  132   // LDS row stride (floats): (m*132 + k) % 64 banks unique per half-wave

// ---------------- kernel 0: zero scratch ----------------
__global__ void k_zero(float* __restrict__ p, long cnt) {
    long i = (long)blockIdx.x * blockDim.x + threadIdx.x;
    long stride = (long)gridDim.x * blockDim.x;
    for (; i < cnt; i += stride) p[i] = 0.0f;
}

// ---------------- kernel 1: edge gather + scatter-add ----------------
// 32 lanes per edge; lane handles 4 consecutive cols (float4, coalesced).
__global__ void k_edge(const float* __restrict__ feat,
                       const int* __restrict__ src,
                       const int* __restrict__ dst,
                       float* __restrict__ agg,
                       float* __restrict__ deg,
                       int E) {
    long tid = (long)blockIdx.x * blockDim.x + threadIdx.x;
    int e = (int)(tid >> 5);
    int lane = (int)(tid & 31);
    if (e >= E) return;
    int s = src[e];
    int d = dst[e];
    const float4 v = *(const float4*)(feat + (long)s * DIM + lane * 4);
    float* a = agg + (long)d * DIM + lane * 4;
    atomicAdd(a + 0, v.x);
    atomicAdd(a + 1, v.y);
    atomicAdd(a + 2, v.z);
    atomicAdd(a + 3, v.w);
    if (lane == 0) atomicAdd(deg + d, 1.0f);
}

// ---------------- kernel 2: mean + GEMM (WMMA f32 16x16x4) + graphnorm + stats ----------------
// Block: 256 threads (8 waves), 64 output rows. Wave w owns 16-col tile, 4 row tiles.
__launch_bounds__(256)
__global__ void k_gemm(const float* __restrict__ feat,
                       const float* __restrict__ agg,
                       const float* __restrict__ deg,
                       const float* __restrict__ W,     // [DIM][DIM], row = out col, col = k
                       const float* __restrict__ bias,  // [DIM]
                       const float* __restrict__ snorm, // [N]
                       float* __restrict__ y,           // [N][DIM]
                       float* __restrict__ colsum,      // [DIM]
                       float* __restrict__ colsumsq,    // [DIM]
                       int N) {
    __shared__ float hs[64 * HS];   // h tile (mean / fallback-to-feature), 64 x 128
    __shared__ float wsh[DIM * HS]; // W staged, 128 x 128 (padded rows)
    __shared__ float sn[64];        // snorm for this row slab

    const int n0 = blockIdx.x * 64;
    const int t  = threadIdx.x;

    // stage W via CDNA5 async global->LDS DMA (b128 per lane, one row per wave iter)
    {
        const int lane4 = (t & 31) * 4;               // float index within row
        const int w0    = t >> 5;                     // wave id
        for (int r = w0; r < DIM; r += 8) {
            unsigned lds_off = (unsigned)(unsigned long long)(void*)&wsh[r * HS + lane4];
            unsigned g_off   = (unsigned)((r * DIM + lane4) * sizeof(float));
            asm volatile("global_load_async_to_lds_b128 %0, %1, %2"
                         :: "v"(lds_off), "v"(g_off), "s"(W)
                         : "memory");
        }
    }
    // stage h rows: deg>0 ? agg/deg : feature ; pad with 0 past N
    for (int i = t; i < 64 * DIM; i += 256) {
        int r = i >> 7, c = i & 127;
        int n = n0 + r;
        float v = 0.0f;
        if (n < N) {
            float dg = deg[n];
            v = (dg > 0.0f) ? agg[(long)n * DIM + c] / dg
                            : feat[(long)n * DIM + c];
        }
        hs[r * HS + c] = v;
    }
    if (t < 64) {
        int n = n0 + t;
        sn[t] = (n < N) ? snorm[n] : 0.0f;
    }
    asm volatile("s_wait_asynccnt 0" ::: "memory");   // own async copies done
    __syncthreads();                                  // all waves' copies visible

    const int w    = t >> 5;        // wave id 0..7 -> column tile
    const int lane = t & 31;
    const int half = lane >> 4;     // 0: K {0,1}; 1: K {2,3} (A/B frag halves)
    const int l16  = lane & 15;
    const int cloc = (w << 4) + l16; // global column 0..127 owned by this lane

    v8f acc[4];
    acc[0] = (v8f){}; acc[1] = (v8f){}; acc[2] = (v8f){}; acc[3] = (v8f){};

    for (int kk = 0; kk < 32; ++kk) {
        const int kb = (kk << 2) + (half << 1);
        // B fragment: B[k][n] = W[n][k]; VGPR0 = K{0|2}, VGPR1 = K{1|3}
        v2f bf;
        bf.x = wsh[cloc * HS + kb];
        bf.y = wsh[cloc * HS + kb + 1];
#pragma unroll
        for (int rt = 0; rt < 4; ++rt) {
            const int m = (rt << 4) + l16;
            v2f af;
            af.x = hs[m * HS + kb];
            af.y = hs[m * HS + kb + 1];
            acc[rt] = __builtin_amdgcn_wmma_f32_16x16x4_f32(
                false, af, false, bf, (short)0, acc[rt], false, false);
        }
    }

    // epilogue: +bias, *snorm, store y, accumulate column stats (lane owns one column)
    const float bi = bias[cloc];
    float s = 0.0f, sq = 0.0f;
    if (n0 + 64 <= N) {
        // fast path: full slab, no per-element guards
#pragma unroll
        for (int rt = 0; rt < 4; ++rt) {
            const int rbase = (rt << 4) + (half << 3); // C layout: vgpr v -> M = v + 8*half
#pragma unroll
            for (int v = 0; v < 8; ++v) {
                const int rloc = rbase + v;
                float val = (acc[rt][v] + bi) * sn[rloc];
                y[(long)(n0 + rloc) * DIM + cloc] = val;
                s  += val;
                sq += val * val;
            }
        }
    } else {
#pragma unroll
        for (int rt = 0; rt < 4; ++rt) {
            const int rbase = (rt << 4) + (half << 3);
#pragma unroll
            for (int v = 0; v < 8; ++v) {
                const int rloc = rbase + v;
                if (n0 + rloc < N) {
                    float val = (acc[rt][v] + bi) * sn[rloc];
                    y[(long)(n0 + rloc) * DIM + cloc] = val;
                    s  += val;
                    sq += val * val;
                }
            }
        }
    }
    atomicAdd(colsum   + cloc, s);
    atomicAdd(colsumsq + cloc, sq);
}

// ---------------- kernel 3: fold BN stats into per-column scale/shift ----------------
__global__ void k_bnprep(const float* __restrict__ colsum,
                         const float* __restrict__ colsumsq,
                         const float* __restrict__ gamma,
                         const float* __restrict__ beta,
                         float* __restrict__ scale,
                         float* __restrict__ shift,
                         int N) {
    int c = threadIdx.x;   // 128 threads
    float invN = 1.0f / (float)N;
    float mu   = colsum[c] * invN;
    float var  = colsumsq[c] * invN - mu * mu;     // biased variance
    float rstd = rsqrtf(var + 1e-5f);
    float sc   = rstd * gamma[c];
    scale[c] = sc;
    shift[c] = beta[c] - mu * sc;
}

// ---------------- kernel 4: batchnorm apply + relu + residual ----------------
__global__ void k_bn(const float* __restrict__ feat,
                     const float* __restrict__ y,
                     const float* __restrict__ scale,
                     const float* __restrict__ shift,
                     float* __restrict__ out,
                     int N) {
    long idx = (long)blockIdx.x * blockDim.x + threadIdx.x; // float4 units
    long total = (long)N * (DIM / 4);
    if (idx >= total) return;
    long n  = idx >> 5;
    int  c4 = (int)(idx & 31) << 2;

    float4 yv = *(const float4*)(y     + n * DIM + c4);
    float4 fv = *(const float4*)(feat  + n * DIM + c4);
    float4 sc = *(const float4*)(scale + c4);
    float4 sh = *(const float4*)(shift + c4);
    float4 ov;
    ov.x = fv.x + fmaxf(yv.x * sc.x + sh.x, 0.0f);
    ov.y = fv.y + fmaxf(yv.y * sc.y + sh.y, 0.0f);
    ov.z = fv.z + fmaxf(yv.z * sc.z + sh.z, 0.0f);
    ov.w = fv.w + fmaxf(yv.w * sc.w + sh.w, 0.0f);
    *(float4*)(out + n * DIM + c4) = ov;
}

// ---------------- launcher ----------------
extern "C" void kernel_launch(void* const* d_in, const int* in_sizes, int n_in,
                              void* d_out, int out_size, void* d_ws, size_t ws_size,
                              hipStream_t stream) {
    const float* feat  = (const float*)d_in[0];
    const float* snorm = (const float*)d_in[1];
    const int*   src   = (const int*)d_in[2];
    const int*   dst   = (const int*)d_in[3];
    const float* W     = (const float*)d_in[4];
    const float* bias  = (const float*)d_in[5];
    const float* gamma = (const float*)d_in[6];
    const float* beta  = (const float*)d_in[7];

    const int N = in_sizes[1];          // snorm_n has N elements
    const int E = in_sizes[2];
    const long ND = (long)N * DIM;

    float* ws       = (float*)d_ws;
    float* agg      = ws;                       // ND
    float* deg      = agg + ND;                 // N
    float* colsum   = deg + N;                  // 128
    float* colsumsq = colsum + DIM;             // 128
    float* scale    = colsumsq + DIM;           // 128
    float* shift    = scale + DIM;              // 128
    float* y        = shift + DIM;              // ND (offset ND+N+512: 16B-aligned)

    float* outp = (float*)d_out;

    // 0) zero agg + deg + stats
    {
        long cnt = ND + N + 2 * DIM;            // agg, deg, colsum, colsumsq
        int blocks = (int)((cnt + 255) / 256);
        k_zero<<<blocks, 256, 0, stream>>>(agg, cnt);
    }
    // 1) edge scatter-add
    {
        long threads = (long)E * 32;
        int blocks = (int)((threads + 255) / 256);
        k_edge<<<blocks, 256, 0, stream>>>(feat, src, dst, agg, deg, E);
    }
    // 2) mean + GEMM (WMMA) + graphnorm + column stats
    {
        int blocks = (N + 63) / 64;
        k_gemm<<<blocks, 256, 0, stream>>>(feat, agg, deg, W, bias, snorm,
                                           y, colsum, colsumsq, N);
    }
    // 3) fold BN stats
    k_bnprep<<<1, DIM, 0, stream>>>(colsum, colsumsq, gamma, beta, scale, shift, N);
    // 4) batchnorm apply + relu + residual
    {
        long threads = (long)N * (DIM / 4);
        int blocks = (int)((threads + 255) / 256);
        k_bn<<<blocks, 256, 0, stream>>>(feat, y, scale, shift, outp, N);
    }
}